// VectorQuantizer_15152644620355
// MI455X (gfx1250) — compile-verified
//
#include <hip/hip_runtime.h>

typedef float v2f __attribute__((ext_vector_type(2)));
typedef float v4f __attribute__((ext_vector_type(4)));
typedef float v8f __attribute__((ext_vector_type(8)));

#define N_TOKENS 1048576
#define EMBED 128
#define NUM_E 21
#define TILE_TOKENS 64
#define THREADS 128
#define ROWP 132 /* padded LDS row stride in floats (128 + 4 kills bank conflicts) */

__global__ __launch_bounds__(THREADS) void vq_main_kernel(
    const float* __restrict__ z_e, const float* __restrict__ emb,
    float* __restrict__ out, float* __restrict__ partial)
{
    __shared__ __align__(16) float sZ[TILE_TOKENS * ROWP];   // 33,792 B
    __shared__ __align__(16) float sE[32 * ROWP];            // 16,896 B
    __shared__ float sZN[TILE_TOKENS];
    __shared__ float sEN[32];
    __shared__ int   sIdx[TILE_TOKENS];
    __shared__ float sPS[THREADS];
    __shared__ float sWL[8];

    const int tid  = threadIdx.x;
    const int lane = tid & 31;
    const int w    = tid >> 5;
    const int tileBase = blockIdx.x * TILE_TOKENS;

    // ---- Stage z_e tile: fully coalesced b128 loads, flat index mapping ----
    // f = i*512 + tid*4  ->  row t = i*4 + w (wave-uniform), col c = lane*4
    #pragma unroll 4
    for (int i = 0; i < 16; ++i) {
        int f = i * (THREADS * 4) + tid * 4;
        int t = f >> 7;
        int c = f & 127;
        v4f v = *reinterpret_cast<const v4f*>(z_e + (size_t)(tileBase + t) * EMBED + c);
        *reinterpret_cast<v4f*>(&sZ[t * ROWP + c]) = v;
    }
    // ---- Stage codebook (21x128), zero-pad rows 21..31 ----
    for (int i = tid; i < NUM_E * 32; i += THREADS) {
        int r = i >> 5;
        int c = (i & 31) * 4;
        v4f v = *reinterpret_cast<const v4f*>(emb + r * EMBED + c);
        *reinterpret_cast<v4f*>(&sE[r * ROWP + c]) = v;
    }
    for (int i = tid; i < (32 - NUM_E) * ROWP; i += THREADS)
        sE[NUM_E * ROWP + i] = 0.0f;
    __syncthreads();

    // ---- ||z||^2 per token (2 threads / token) ----
    {
        int t = tid >> 1, h = tid & 1;
        float s = 0.f;
        const float* row = &sZ[t * ROWP + h * 64];
        #pragma unroll 4
        for (int i = 0; i < 64; i += 4) {
            v4f v = *reinterpret_cast<const v4f*>(row + i);
            s += v[0]*v[0] + v[1]*v[1] + v[2]*v[2] + v[3]*v[3];
        }
        sPS[tid] = s;
    }
    __syncthreads();
    if (tid < TILE_TOKENS) sZN[tid] = sPS[2 * tid] + sPS[2 * tid + 1];
    // ---- ||e||^2 per code (pad rows get +inf sentinel -> never argmin) ----
    if (tid < 32) {
        float s = 0.f;
        for (int i = 0; i < EMBED; ++i) { float x = sE[tid * ROWP + i]; s += x * x; }
        sEN[tid] = (tid < NUM_E) ? s : 1e30f;
    }
    __syncthreads();

    // ---- WMMA: dots[16 tokens x 32 codes] in fp32, K=128 as 32 chunks of 4 ----
    // A 16x4 f32 layout: lane M = l&15, K = (l>>4)*2 + vgpr.  B 4x16 symmetric.
    const int mrow = lane & 15;
    const int koff = (lane >> 4) * 2;
    const int waveBase = w * 16;
    v8f c0 = {0,0,0,0,0,0,0,0};
    v8f c1 = {0,0,0,0,0,0,0,0};
    const float* aRow  = &sZ[(waveBase + mrow) * ROWP + koff];
    const float* bRow0 = &sE[mrow * ROWP + koff];
    const float* bRow1 = &sE[(16 + mrow) * ROWP + koff];
    #pragma unroll
    for (int kk = 0; kk < 32; ++kk) {
        v2f a  = *reinterpret_cast<const v2f*>(aRow  + kk * 4);
        v2f b0 = *reinterpret_cast<const v2f*>(bRow0 + kk * 4);
        v2f b1 = *reinterpret_cast<const v2f*>(bRow1 + kk * 4);
        c0 = __builtin_amdgcn_wmma_f32_16x16x4_f32(false, a, false, b0, (short)0, c0, false, false);
        c1 = __builtin_amdgcn_wmma_f32_16x16x4_f32(false, a, false, b1, (short)0, c1, false, false);
    }

    // ---- argmin over 21 codes per token; C layout: M = r + (l>=16)*8, N = l&15 ----
    float lossAcc = 0.f;
    const float en0 = sEN[mrow];
    const float en1 = sEN[16 + mrow];
    #pragma unroll
    for (int r = 0; r < 8; ++r) {
        float v0 = en0 - 2.f * c0[r];          // candidate j = mrow (always < 21)
        float v1 = en1 - 2.f * c1[r];          // candidate j = 16+mrow (1e30 if padded)
        float v; int j;
        if (v1 < v0) { v = v1; j = 16 + mrow; } else { v = v0; j = mrow; }
        #pragma unroll
        for (int m = 1; m <= 8; m <<= 1) {     // butterfly within each 16-lane half
            float ov = __shfl_xor(v, m, 32);
            int   oj = __shfl_xor(j, m, 32);
            if (ov < v || (ov == v && oj < j)) { v = ov; j = oj; }
        }
        if (mrow == 0) {
            int tk = waveBase + r + (lane >> 4) * 8;
            sIdx[tk] = j;
            lossAcc += sZN[tk] + v;            // = min squared distance for token tk
        }
    }
    if (mrow == 0) sWL[w * 2 + (lane >> 4)] = lossAcc;
    __syncthreads();

    // ---- Write z_q rows: index uniform per wave-iteration -> coalesced b128 ----
    #pragma unroll 4
    for (int i = 0; i < 16; ++i) {
        int t = i * 4 + w;
        int j = sIdx[t];
        int c = lane * 4;
        v4f v = *reinterpret_cast<const v4f*>(&sE[j * ROWP + c]);
        *reinterpret_cast<v4f*>(out + (size_t)(tileBase + t) * EMBED + c) = v;
    }
    // ---- indices (as float) after z_q (N*128 elems) and loss (1 elem) ----
    if (tid < TILE_TOKENS)
        out[(size_t)N_TOKENS * EMBED + 1 + tileBase + tid] = (float)sIdx[tid];
    if (tid == 0) {
        float s = 0.f;
        #pragma unroll
        for (int i = 0; i < 8; ++i) s += sWL[i];
        partial[blockIdx.x] = s;
    }
}

__global__ __launch_bounds__(256) void vq_reduce_kernel(
    const float* __restrict__ partial, float* __restrict__ out)
{
    __shared__ float sred[256];
    const int nblk = N_TOKENS / TILE_TOKENS;
    float s = 0.f;
    for (int i = threadIdx.x; i < nblk; i += 256) s += partial[i];
    sred[threadIdx.x] = s;
    __syncthreads();
    for (int off = 128; off > 0; off >>= 1) {
        if ((int)threadIdx.x < off) sred[threadIdx.x] += sred[threadIdx.x + off];
        __syncthreads();
    }
    if (threadIdx.x == 0)
        out[(size_t)N_TOKENS * EMBED] =
            sred[0] * ((1.0f + 0.1f) / ((float)N_TOKENS * (float)EMBED));
}

extern "C" void kernel_launch(void* const* d_in, const int* in_sizes, int n_in,
                              void* d_out, int out_size, void* d_ws, size_t ws_size,
                              hipStream_t stream) {
    const float* z_e = (const float*)d_in[0];
    const float* emb = (const float*)d_in[1];
    float* out = (float*)d_out;
    float* partial = (float*)d_ws;   // nblk * 4 bytes = 64 KB
    const int nblk = N_TOKENS / TILE_TOKENS;   // 16384
    vq_main_kernel<<<nblk, THREADS, 0, stream>>>(z_e, emb, out, partial);
    vq_reduce_kernel<<<1, 256, 0, stream>>>(partial, out);
    (void)in_sizes; (void)n_in; (void)out_size; (void)ws_size;
}